// ApsUp_420906795252
// MI455X (gfx1250) — compile-verified
//
#include <hip/hip_runtime.h>
#include <cstdint>

// B=16, C=256, N=64 -> out [16,256,128,128] f32.
// Memory-bound (67MB in + 268MB out => ~14.4us floor @ 23.3TB/s).
// Strategy: per-(b,ch) block stages the 16KB channel image in LDS via
// CDNA5 async global->LDS b128 loads (ASYNCcnt), computes each output 2x2
// cell from 4 LDS values, and streams results with non-temporal b64 stores.

#define B_ 16
#define C_ 256
#define N_ 64

__global__ __launch_bounds__(256) void apsup_blur_kernel(
    const float* __restrict__ inp,
    const int*   __restrict__ phases,
    float*       __restrict__ out)
{
    __shared__ float tile[N_ * N_];   // 16 KB

    const int bc = blockIdx.x;        // b*C + ch
    const int b  = bc >> 8;
    const int t  = threadIdx.x;

    const int p = phases[b];          // phase = r + 2*c
    const int r = p & 1;
    const int c = (p >> 1) & 1;

    // ---- async copy 16KB global -> LDS: 4 x b128 per thread ----
    const unsigned long long srcBase =
        (unsigned long long)(uintptr_t)(inp) + (unsigned long long)bc * (N_ * N_ * 4u);
    // low 32 bits of generic shared pointer == workgroup-relative LDS byte offset
    const unsigned ldsBase = (unsigned)(uintptr_t)(&tile[0]);
#pragma unroll
    for (int it = 0; it < 4; ++it) {
        const int k = t + it * 256;                       // 16-byte chunk id
        const unsigned ldsAddr = ldsBase + (unsigned)(k * 16);
        const unsigned long long gaddr = srcBase + (unsigned long long)(k * 16);
        asm volatile("global_load_async_to_lds_b128 %0, %1, off"
                     :: "v"(ldsAddr), "v"(gaddr)
                     : "memory");
    }
    asm volatile("s_wait_asynccnt 0" ::: "memory");
    __syncthreads();

    // ---- compute: 64x64 cells, each produces a 2x2 output block ----
    const int ru = (r == 0) ? 1 : (N_ - 1);   // row neighbor step (mod 64)
    const int cu = (c == 0) ? 1 : (N_ - 1);   // col neighbor step (mod 64)
    float* dst = out + (size_t)bc * (2 * N_ * 2 * N_);

#pragma unroll 4
    for (int pass = 0; pass < 16; ++pass) {
        const int m  = pass * 256 + t;        // cell id, lanes -> consecutive v
        const int u  = m >> 6;
        const int v  = m & (N_ - 1);
        const int rb = (u + ru) & (N_ - 1);
        const int cb = (v + cu) & (N_ - 1);

        const float x00 = tile[(u  << 6) | v ];
        const float x01 = tile[(u  << 6) | cb];
        const float x10 = tile[(rb << 6) | v ];
        const float x11 = tile[(rb << 6) | cb];

        const float o_cc = 0.25f   *  x00;                       // center tap
        const float o_ch = 0.125f  * (x00 + x01);                // horiz edge
        const float o_hc = 0.125f  * (x00 + x10);                // vert edge
        const float o_hh = 0.0625f * (x00 + x01 + x10 + x11);    // corner

        // row 2u holds parity-s==0; input row parity is r.
        const float a_c = (r == 0) ? o_cc : o_hc;   // row 2u,   col parity == c
        const float a_n = (r == 0) ? o_ch : o_hh;   // row 2u,   other col
        const float b_c = (r == 0) ? o_hc : o_cc;   // row 2u+1, col parity == c
        const float b_n = (r == 0) ? o_hh : o_ch;   // row 2u+1, other col

        union { float f[2]; unsigned long long q; } pa, pb;
        pa.f[0] = (c == 0) ? a_c : a_n;  pa.f[1] = (c == 0) ? a_n : a_c;
        pb.f[0] = (c == 0) ? b_c : b_n;  pb.f[1] = (c == 0) ? b_n : b_c;

        const size_t row0 = (size_t)(2 * u) * (2 * N_) + (size_t)(2 * v);
        __builtin_nontemporal_store(pa.q, (unsigned long long*)(dst + row0));
        __builtin_nontemporal_store(pb.q, (unsigned long long*)(dst + row0 + 2 * N_));
    }
}

extern "C" void kernel_launch(void* const* d_in, const int* in_sizes, int n_in,
                              void* d_out, int out_size, void* d_ws, size_t ws_size,
                              hipStream_t stream) {
    const float* inp    = (const float*)d_in[0];
    const int*   phases = (const int*)d_in[1];
    float*       out    = (float*)d_out;
    (void)in_sizes; (void)n_in; (void)out_size; (void)d_ws; (void)ws_size;

    apsup_blur_kernel<<<B_ * C_, 256, 0, stream>>>(inp, phases, out);
}